// SnakeSimpleLoss_7344394076749
// MI455X (gfx1250) — compile-verified
//
#include <hip/hip_runtime.h>
#include <math.h>

// ---- problem constants (match reference) ----
constexpr int   Bn = 4, Hh = 256, Ww = 256, Nn = 128;
constexpr int   KR = 4, KS = 9;                 // gauss filter radius / size
constexpr float STEPSZ = 0.1f, ALPHA_C = 0.01f, BETA_C = 0.01f;
constexpr int   NSTEPS = 50;
constexpr float DMAX = 15.0f, EXTGRADFAC = 10.0f, ENHANCE = 2.0f;

typedef __attribute__((ext_vector_type(2))) float v2f;
typedef __attribute__((ext_vector_type(8))) float v8f;

#if __has_builtin(__builtin_amdgcn_sqrtf)
#define FAST_SQRT(x) __builtin_amdgcn_sqrtf(x)   // single v_sqrt_f32
#else
#define FAST_SQRT(x) sqrtf(x)
#endif

// ------------------------------------------------------------------
// 1) gimg = conv2d(pred, fltr, SAME, zero-pad) * EXTGRADFAC
//    fltr: (2,1,9,9); one thread per pixel computes both channels.
// ------------------------------------------------------------------
__global__ void conv_grad_kernel(const float* __restrict__ pred,
                                 const float* __restrict__ fltr,
                                 float* __restrict__ gimg) {
    __shared__ float f[2 * KS * KS];
    int t = threadIdx.x;
    if (t < 2 * KS * KS) f[t] = fltr[t];
    __syncthreads();

    int idx = blockIdx.x * blockDim.x + t;
    if (idx >= Bn * Hh * Ww) return;
    int b   = idx / (Hh * Ww);
    int rem = idx % (Hh * Ww);
    int y   = rem / Ww;
    int x   = rem % Ww;
    const float* p = pred + (size_t)b * Hh * Ww;

    float s0 = 0.f, s1 = 0.f;
    #pragma unroll
    for (int i = 0; i < KS; ++i) {
        int yy = y + i - KR;
        if (yy < 0 || yy >= Hh) continue;
        #pragma unroll
        for (int j = 0; j < KS; ++j) {
            int xx = x + j - KR;
            if (xx < 0 || xx >= Ww) continue;
            float v = p[yy * Ww + xx];
            s0 = fmaf(v, f[i * KS + j], s0);
            s1 = fmaf(v, f[KS * KS + i * KS + j], s1);
        }
    }
    gimg[(size_t)b * 2 * Hh * Ww + y * Ww + x]            = s0 * EXTGRADFAC;
    gimg[(size_t)b * 2 * Hh * Ww + Hh * Ww + y * Ww + x]  = s1 * EXTGRADFAC;
}

// ------------------------------------------------------------------
// 2) 50-step snake evolution. One block per batch, one thread per node,
//    node state in LDS (this path is serial; LDS keeps it on-WGP).
// ------------------------------------------------------------------
__global__ void snake_kernel(const float* __restrict__ nodes_in,
                             const float* __restrict__ gimg,
                             float* __restrict__ nodes_out) {
    __shared__ float xr[Nn], xc[Nn], d2r[Nn], d2c[Nn];
    int b = blockIdx.x;
    int i = threadIdx.x;
    xr[i] = nodes_in[((size_t)b * Nn + i) * 2 + 0];
    xc[i] = nodes_in[((size_t)b * Nn + i) * 2 + 1];
    const float* g0 = gimg + (size_t)b * 2 * Hh * Ww;
    const float* g1 = g0 + Hh * Ww;
    int im = i > 0 ? i - 1 : 0;
    int ip = i < Nn - 1 ? i + 1 : Nn - 1;

    for (int s = 0; s < NSTEPS; ++s) {
        __syncthreads();
        float r = xr[i], c = xc[i];
        float dr = xr[im] - 2.f * r + xr[ip];
        float dc = xc[im] - 2.f * c + xc[ip];

        // bilinear sample of both gradient channels at (r,c)
        float rr = fminf(fmaxf(r, 0.f), (float)(Hh - 1));
        float cc = fminf(fmaxf(c, 0.f), (float)(Ww - 1));
        int r0 = (int)floorf(rr), c0 = (int)floorf(cc);
        int r1 = min(r0 + 1, Hh - 1), c1 = min(c0 + 1, Ww - 1);
        float fr = rr - (float)r0, fc = cc - (float)c0;
        float w00 = (1.f - fr) * (1.f - fc), w01 = (1.f - fr) * fc;
        float w10 = fr * (1.f - fc),         w11 = fr * fc;
        float e0 = g0[r0 * Ww + c0] * w00 + g0[r0 * Ww + c1] * w01 +
                   g0[r1 * Ww + c0] * w10 + g0[r1 * Ww + c1] * w11;
        float e1 = g1[r0 * Ww + c0] * w00 + g1[r0 * Ww + c1] * w01 +
                   g1[r1 * Ww + c0] * w10 + g1[r1 * Ww + c1] * w11;

        d2r[i] = dr; d2c[i] = dc;
        __syncthreads();
        float d4r = d2r[im] - 2.f * dr + d2r[ip];
        float d4c = d2c[im] - 2.f * dc + d2c[ip];
        float fR = ALPHA_C * dr - BETA_C * d4r + e0;
        float fC = ALPHA_C * dc - BETA_C * d4c + e1;
        __syncthreads();
        xr[i] = r + STEPSZ * fR;
        xc[i] = c + STEPSZ * fC;
    }
    __syncthreads();
    nodes_out[((size_t)b * Nn + i) * 2 + 0] = xr[i];
    nodes_out[((size_t)b * Nn + i) * 2 + 1] = xc[i];
}

// ------------------------------------------------------------------
// 3) render_dmap + MSE via V_WMMA_F32_16X16X4_F32.
//    Full squared distance folded into the rank-4 product:
//      A row m = (y, x_m, 1, y^2 + x_m^2)
//      B col n = (-2*nr, -2*nc, nr^2 + nc^2, 1)
//      C[m,n]  = (y - nr)^2 + (x_m - nc)^2     (exact, f32)
//    One wave = 16-pixel row tile; 8 unrolled WMMAs cover 128 nodes;
//    butterfly min within 16-lane halves (matches C/D lane layout).
// ------------------------------------------------------------------
__global__ void render_loss_kernel(const float* __restrict__ pred,
                                   const float* __restrict__ nodes,
                                   const float* __restrict__ widths,
                                   float* __restrict__ acc) {
    const int lane = threadIdx.x & 31;
    const int wave = threadIdx.x >> 5;
    const int tile = blockIdx.x * (blockDim.x >> 5) + wave;   // 0 .. B*H*W/16-1
    const int tilesPerBatch = Hh * (Ww / 16);                 // 4096 (pow2)
    const int b     = tile / tilesPerBatch;
    const int rem   = tile % tilesPerBatch;
    const int y     = rem / (Ww / 16);
    const int xbase = (rem % (Ww / 16)) * 16;
    const int half  = lane >> 4;         // 0: holds K=0,1 ; 1: holds K=2,3
    const int l16   = lane & 15;

    const float yf = (float)y;
    const float xm = (float)(xbase + l16);   // x of pixel row M=l16

    // A (16x4 f32): lanes<16 carry (K0,K1)=(y, x_m); lanes>=16 carry
    // (K2,K3)=(1, y^2+x_m^2) for the same row m = lane-16.
    v2f a;
    a.x = half ? 1.0f : yf;
    a.y = half ? (yf * yf + xm * xm) : xm;

    float m[8];
    #pragma unroll
    for (int k = 0; k < 8; ++k) m[k] = 3.0e38f;

    const float* nb = nodes  + (size_t)b * Nn * 2;
    const float* wb = widths + (size_t)b * Nn;

    #pragma unroll
    for (int nt = 0; nt < Nn / 16; ++nt) {
        int   n  = nt * 16 + l16;
        float nr = nb[2 * n + 0];
        float nc = nb[2 * n + 1];
        float wn = wb[n];
        // B (4x16 f32): col n = (-2nr, -2nc, nr^2+nc^2, 1)
        v2f bv;
        bv.x = half ? (nr * nr + nc * nc) : (-2.0f * nr);
        bv.y = half ? 1.0f                : (-2.0f * nc);
        v8f c = {};
        c = __builtin_amdgcn_wmma_f32_16x16x4_f32(
                /*neg_a=*/false, a, /*neg_b=*/false, bv,
                /*c_mod=*/(short)0, c, /*reuse_a=*/false, /*reuse_b=*/false);
        #pragma unroll
        for (int k = 0; k < 8; ++k) {
            float d = FAST_SQRT(fmaxf(c[k], 0.0f)) - wn;
            m[k] = fminf(m[k], d);
        }
    }
    // min across the 16 node-columns: butterfly stays inside each 16-lane half
    #pragma unroll
    for (int off = 1; off < 16; off <<= 1) {
        #pragma unroll
        for (int k = 0; k < 8; ++k)
            m[k] = fminf(m[k], __shfl_xor(m[k], off, 32));
    }
    // lane 0 owns pixels 0..7, lane 16 owns pixels 8..15
    if (l16 == 0) {
        float lacc = 0.0f;
        #pragma unroll
        for (int k = 0; k < 8; ++k) {
            float d = fminf(fmaxf(m[k], -DMAX), DMAX);
            if (d < 0.f) d *= ENHANCE;
            d = fminf(fmaxf(d, -DMAX), DMAX);
            int   px = xbase + k + 8 * half;
            float e  = pred[(size_t)b * Hh * Ww + y * Ww + px] - d;
            lacc = fmaf(e, e, lacc);
        }
        atomicAdd(acc, lacc);
    }
}

__global__ void finalize_kernel(const float* __restrict__ acc,
                                float* __restrict__ out) {
    if (threadIdx.x == 0 && blockIdx.x == 0)
        out[0] = acc[0] / (float)(Bn * Hh * Ww);
}

// ------------------------------------------------------------------
extern "C" void kernel_launch(void* const* d_in, const int* in_sizes, int n_in,
                              void* d_out, int out_size, void* d_ws, size_t ws_size,
                              hipStream_t stream) {
    const float* pred   = (const float*)d_in[0];   // (B,1,H,W)
    const float* nodes  = (const float*)d_in[1];   // (B,N,2)
    const float* widths = (const float*)d_in[2];   // (B,N)
    const float* fltr   = (const float*)d_in[3];   // (2,1,9,9)
    float* out = (float*)d_out;

    float* gimg    = (float*)d_ws;                         // B*2*H*W floats (2 MB)
    float* nodes_f = gimg + (size_t)Bn * 2 * Hh * Ww;      // B*N*2 floats
    float* acc     = nodes_f + (size_t)Bn * Nn * 2;        // 1 float

    hipMemsetAsync(acc, 0, sizeof(float), stream);

    conv_grad_kernel<<<(Bn * Hh * Ww + 255) / 256, 256, 0, stream>>>(pred, fltr, gimg);
    snake_kernel<<<Bn, Nn, 0, stream>>>(nodes, gimg, nodes_f);
    // tiles = B*H*(W/16) = 16384; 8 waves (tiles) per 256-thread block
    render_loss_kernel<<<(Bn * Hh * (Ww / 16)) / 8, 256, 0, stream>>>(pred, nodes_f, widths, acc);
    finalize_kernel<<<1, 1, 0, stream>>>(acc, out);
}